// NoiseLinear_34943853920604
// MI455X (gfx1250) — compile-verified
//
#include <hip/hip_runtime.h>
#include <hip/hip_bf16.h>
#include <stdint.h>

// ---------------------------------------------------------------------------
// Problem: out[b,o] = x@W^T + bias + 0.1 * sum_i eps[b,o,i]*W[o,i]*x[b,i]
//   B=64, IN=1024, OUT=4096, eps = threefry-normal(key(seed), B*OUT*IN)
// ---------------------------------------------------------------------------

#define B_DIM   64
#define IN_DIM  1024
#define OUT_DIM 4096
#define NOISE   0.1f

typedef __attribute__((ext_vector_type(2))) float v2f;
typedef __attribute__((ext_vector_type(8))) float v8f;

// ======================== Kernel 1: WMMA f32 GEMM ==========================
// lin = x @ W^T + bias, one wave per 16x16 tile, V_WMMA_F32_16X16X4_F32.
//
// 32-bit A (16x4, MxK) layout: lanes 0-15 -> M=lane, K={0,1} in v0,v1;
//                              lanes 16-31 -> M=lane-16, K={2,3}.
// 32-bit B (4x16, KxN) mirrors it with N across lanes.
// 32-bit C/D (16x16): VGPR r holds M=r (lanes 0-15) and M=r+8 (lanes 16-31).
__global__ __launch_bounds__(256) void noiselinear_gemm_wmma(
    const float* __restrict__ x,      // [64,1024]
    const float* __restrict__ w,      // [4096,1024]
    const float* __restrict__ bias,   // [4096]
    float* __restrict__ out) {        // [64,4096]
  const int wave = (int)((blockIdx.x * blockDim.x + threadIdx.x) >> 5);
  const int lane = (int)(threadIdx.x & 31);

  const int row_tile = wave >> 8;    // 0..3   (64/16)
  const int col_tile = wave & 255;   // 0..255 (4096/16)

  const int half = lane >> 4;        // 0 or 1
  const int l15  = lane & 15;

  const int m = row_tile * 16 + l15; // A row this lane feeds
  const int n = col_tile * 16 + l15; // B col this lane feeds

  const float* xp = x + (size_t)m * IN_DIM + half * 2; // K pair base for A
  const float* wp = w + (size_t)n * IN_DIM + half * 2; // K pair base for B

  v8f c = {};
#pragma unroll 8
  for (int k = 0; k < IN_DIM; k += 4) {
    v2f a = *reinterpret_cast<const v2f*>(xp + k);  // global_load_b64
    v2f b = *reinterpret_cast<const v2f*>(wp + k);  // global_load_b64
    // 8 args: (neg_a, A, neg_b, B, c_mod, C, reuse_a, reuse_b)
    c = __builtin_amdgcn_wmma_f32_16x16x4_f32(
        false, a, false, b, (short)0, c, false, false);
  }

  const float bn = bias[n];
#pragma unroll
  for (int r = 0; r < 8; ++r) {
    const int mm = row_tile * 16 + half * 8 + r;
    out[(size_t)mm * OUT_DIM + n] = c[r] + bn;
  }
}

// ===================== Kernel 2: threefry noise term =======================

__device__ __forceinline__ uint32_t rotl32(uint32_t v, int r) {
  return (v << r) | (v >> (32 - r));
}

// Exact JAX threefry2x32: 20 rounds, key injection every 4.
__device__ __forceinline__ void threefry2x32(uint32_t ks0, uint32_t ks1,
                                             uint32_t c0, uint32_t c1,
                                             uint32_t& o0, uint32_t& o1) {
  const uint32_t ks2 = ks0 ^ ks1 ^ 0x1BD11BDAu;
  uint32_t x0 = c0 + ks0;
  uint32_t x1 = c1 + ks1;

#define TF_ROUND(r) { x0 += x1; x1 = rotl32(x1, r); x1 ^= x0; }
  TF_ROUND(13) TF_ROUND(15) TF_ROUND(26) TF_ROUND(6)
  x0 += ks1; x1 += ks2 + 1u;
  TF_ROUND(17) TF_ROUND(29) TF_ROUND(16) TF_ROUND(24)
  x0 += ks2; x1 += ks0 + 2u;
  TF_ROUND(13) TF_ROUND(15) TF_ROUND(26) TF_ROUND(6)
  x0 += ks0; x1 += ks1 + 3u;
  TF_ROUND(17) TF_ROUND(29) TF_ROUND(16) TF_ROUND(24)
  x0 += ks1; x1 += ks2 + 4u;
  TF_ROUND(13) TF_ROUND(15) TF_ROUND(26) TF_ROUND(6)
  x0 += ks2; x1 += ks0 + 5u;
#undef TF_ROUND
  o0 = x0;
  o1 = x1;
}

// JAX uniform(-0.99999994, 1) via mantissa trick, then sqrt(2)*erfinv(u).
__device__ __forceinline__ float bits_to_normal(uint32_t bits) {
  const float lo = -0.99999994f;                  // nextafter(-1, 0)
  float f = __uint_as_float(0x3F800000u | (bits >> 9)) - 1.0f; // [0,1)
  float u = fmaf(f, 1.0f - lo, lo);
  u = fmaxf(u, lo);
  return 1.41421356237f * erfinvf(u);
}

// One wave handles the pair of outputs (b, o) and (b+32, o), b in [0,32):
// element n = ((b*OUT+o)*IN + i) < 2^27 pairs with n + 2^27 which is the
// identical (o,i) at batch row b+32 -> one threefry hash feeds both rows.
__global__ __launch_bounds__(256) void noiselinear_noise(
    const float* __restrict__ x,      // [64,1024]
    const float* __restrict__ w,      // [4096,1024]
    const int*   __restrict__ seedp,  // [1]
    float* __restrict__ out) {        // [64,4096] (pre-filled by GEMM)
  const int wave = (int)((blockIdx.x * blockDim.x + threadIdx.x) >> 5);
  const int lane = (int)(threadIdx.x & 31);

  const int b = wave >> 12;           // 0..31
  const int o = wave & (OUT_DIM - 1); // 0..4095

  // jax.random.key(seed): key data = (hi32, lo32) of the seed.
  const uint32_t ks0 = 0u;
  const uint32_t ks1 = (uint32_t)seedp[0];

  const uint32_t base = (uint32_t)(((uint32_t)b * OUT_DIM + o) * IN_DIM);
  const uint32_t HALF = 1u << 27;     // (B*OUT*IN)/2

  const float* __restrict__ x0p = x + (size_t)b * IN_DIM;
  const float* __restrict__ x1p = x + (size_t)(b + 32) * IN_DIM;
  const float* __restrict__ wp  = w + (size_t)o * IN_DIM;

  float acc0 = 0.0f;
  float acc1 = 0.0f;

  for (int j = 0; j < IN_DIM; j += 32) {
    const int i = j + lane;           // coalesced across the wave
    uint32_t r0, r1;
    threefry2x32(ks0, ks1, base + (uint32_t)i, base + (uint32_t)i + HALF,
                 r0, r1);
    const float e0 = bits_to_normal(r0);  // eps[b,    o, i]
    const float e1 = bits_to_normal(r1);  // eps[b+32, o, i]
    const float wv = wp[i];
    acc0 = fmaf(e0 * wv, x0p[i], acc0);
    acc1 = fmaf(e1 * wv, x1p[i], acc1);
  }

  // wave32 butterfly reduction
#pragma unroll
  for (int off = 16; off > 0; off >>= 1) {
    acc0 += __shfl_xor(acc0, off, 32);
    acc1 += __shfl_xor(acc1, off, 32);
  }

  if (lane == 0) {
    out[(size_t)b * OUT_DIM + o]        += NOISE * acc0;
    out[(size_t)(b + 32) * OUT_DIM + o] += NOISE * acc1;
  }
}

// =============================== Launcher ==================================
extern "C" void kernel_launch(void* const* d_in, const int* in_sizes, int n_in,
                              void* d_out, int out_size, void* d_ws,
                              size_t ws_size, hipStream_t stream) {
  const float* x    = (const float*)d_in[0];
  const float* w    = (const float*)d_in[1];
  const float* bias = (const float*)d_in[2];
  const int*   seed = (const int*)d_in[3];
  float* out = (float*)d_out;

  // GEMM: (64/16)*(4096/16) = 1024 tiles -> 1024 waves -> 128 blocks of 256.
  noiselinear_gemm_wmma<<<128, 256, 0, stream>>>(x, w, bias, out);

  // Noise: 32*4096 = 131072 wave-pairs -> 16384 blocks of 256 (8 waves each).
  noiselinear_noise<<<16384, 256, 0, stream>>>(x, w, seed, out);
}